// GGCARAFE_19078244729006
// MI455X (gfx1250) — compile-verified
//
#include <hip/hip_runtime.h>

typedef float v2f __attribute__((ext_vector_type(2)));
typedef float v8f __attribute__((ext_vector_type(8)));

#define CIN   256
#define CCOMP 64
#define CENC  36
#define HH    64
#define WWID  64
#define HW    4096          // 64*64
#define KK    9
#define SS    4
#define KDIM  576           // CCOMP * 9

// ---------------------------------------------------------------------------
// K1: channel mean  mean[b,h,w] = (1/256) sum_c x[b,c,h,w]
// consecutive threads -> consecutive w => coalesced at every c step
// ---------------------------------------------------------------------------
__global__ void gg_mean_kernel(const float* __restrict__ x,
                               float* __restrict__ mean) {
    int idx = blockIdx.x * 256 + threadIdx.x;     // 0 .. 32767
    int b   = idx >> 12;                          // /4096
    int pix = idx & 4095;
    const float* xb = x + (size_t)b * CIN * HW + pix;
    float s = 0.f;
#pragma unroll 8
    for (int c = 0; c < CIN; ++c) s += xb[c * HW];
    mean[idx] = s * (1.0f / CIN);
}

// ---------------------------------------------------------------------------
// K2: comp = w1[64x256] @ x[b][256x4096]  via V_WMMA_F32_16X16X4_F32
// block = 256 threads = 8 waves; wave -> (mTile = wave%4, nTile = wave/4)
// grid = B * (4096/32) = 1024 blocks
// ---------------------------------------------------------------------------
__global__ void gg_comp_gemm(const float* __restrict__ x,
                             const float* __restrict__ w1,
                             const float* __restrict__ b1,
                             float* __restrict__ comp) {
    int b     = blockIdx.x >> 7;            // / 128
    int nBase = (blockIdx.x & 127) * 32;
    int wave  = threadIdx.x >> 5;
    int lane  = threadIdx.x & 31;
    int lr    = lane & 15;
    int half  = lane >> 4;
    int m0    = (wave & 3) * 16;
    int n0    = nBase + (wave >> 2) * 16;

    const float* xb   = x + (size_t)b * CIN * HW;
    const float* aRow = w1 + (m0 + lr) * CIN;

    v8f acc = {};
    for (int k = 0; k < CIN; k += 4) {
        int ke = k + 2 * half;              // lanes 0-15: K=k,k+1 ; 16-31: K=k+2,k+3
        v2f a, bm;
        a.x  = aRow[ke];
        a.y  = aRow[ke + 1];
        bm.x = xb[ke * HW + n0 + lr];
        bm.y = xb[(ke + 1) * HW + n0 + lr];
        acc = __builtin_amdgcn_wmma_f32_16x16x4_f32(
                  false, a, false, bm, (short)0, acc, false, false);
    }

    float* cb = comp + (size_t)b * CCOMP * HW;
#pragma unroll
    for (int r = 0; r < 8; ++r) {
        int m = m0 + r + 8 * half;          // C layout: VGPR r -> M=r / M=r+8
        cb[m * HW + n0 + lr] = acc[r] + b1[m];
    }
}

// ---------------------------------------------------------------------------
// K3: kern = w2[36x576] @ im2col(comp)[576x4096]  (implicit GEMM, WMMA f32)
// M padded 36 -> 48: 3 waves/block, one 16-row tile each; zero rows m>=36.
// grid = B * 256 nTiles = 2048 blocks of 96 threads
// ---------------------------------------------------------------------------
__global__ void gg_kern_gemm(const float* __restrict__ comp,
                             const float* __restrict__ w2,
                             const float* __restrict__ b2,
                             float* __restrict__ kern) {
    int b    = blockIdx.x >> 8;             // / 256
    int n0   = (blockIdx.x & 255) * 16;
    int wave = threadIdx.x >> 5;            // 0..2
    int lane = threadIdx.x & 31;
    int lr   = lane & 15;
    int half = lane >> 4;
    int m0   = wave * 16;

    int n = n0 + lr;
    int h = n >> 6, w = n & 63;
    const float* cb = comp + (size_t)b * CCOMP * HW;
    int mA = m0 + lr;
    bool aValid = (mA < CENC);

    v8f acc = {};
    for (int k = 0; k < KDIM; k += 4) {
        int ke = k + 2 * half;
        v2f a, bm;
        a.x = aValid ? w2[mA * KDIM + ke]     : 0.f;
        a.y = aValid ? w2[mA * KDIM + ke + 1] : 0.f;
        {   // im2col element for K = ke   (k = c*9 + ky*3 + kx, OIHW order)
            int ci = ke / 9, r = ke - ci * 9, ky = r / 3, kx = r - ky * 3;
            int hh = h + ky - 1, ww = w + kx - 1;
            bm.x = ((unsigned)hh < 64u && (unsigned)ww < 64u)
                       ? cb[ci * HW + hh * 64 + ww] : 0.f;
        }
        {   // im2col element for K = ke+1
            int k1 = ke + 1;
            int ci = k1 / 9, r = k1 - ci * 9, ky = r / 3, kx = r - ky * 3;
            int hh = h + ky - 1, ww = w + kx - 1;
            bm.y = ((unsigned)hh < 64u && (unsigned)ww < 64u)
                       ? cb[ci * HW + hh * 64 + ww] : 0.f;
        }
        acc = __builtin_amdgcn_wmma_f32_16x16x4_f32(
                  false, a, false, bm, (short)0, acc, false, false);
    }

    float* kb = kern + (size_t)b * CENC * HW;
#pragma unroll
    for (int r = 0; r < 8; ++r) {
        int m = m0 + r + 8 * half;
        if (m < CENC) kb[m * HW + n] = acc[r] + b2[m];
    }
}

// ---------------------------------------------------------------------------
// K4: fused grad-guidance + mask + softmax(9) + CARAFE reassembly.
// block = one (b,h) row: 256 thr = 64 w-lanes x 4 channel groups.
// masks live in LDS (9*4*64 floats = 9 KB); x patch reads hit L2 (x fits in
// the 192 MB L2), output writes are the dominant 134 MB HBM stream.
// ---------------------------------------------------------------------------
__global__ void gg_carafe_kernel(const float* __restrict__ x,
                                 const float* __restrict__ mean,
                                 const float* __restrict__ kern,
                                 float* __restrict__ out) {
    __shared__ float smask[KK][SS][WWID];

    int bh = blockIdx.x;                    // 0 .. 511
    int b  = bh >> 6, h = bh & 63;
    int tx = threadIdx.x & 63;              // w
    int ty = threadIdx.x >> 6;              // 0..3

    const float* meanb = mean + b * HW;
    const float* kb    = kern + (size_t)b * CENC * HW;

    if (ty == 0) {
        int w   = tx;
        float mc = meanb[h * 64 + w];
        float mval[SS][KK];
#pragma unroll
        for (int kk = 0; kk < KK; ++kk) {
            int dy = kk / 3 - 1, dx = kk % 3 - 1;
            int hh = h + dy, ww = w + dx;
            float mp = ((unsigned)hh < 64u && (unsigned)ww < 64u)
                           ? meanb[hh * 64 + ww] : 0.f;     // zero-pad patch
            float g = mp - mc;
            g = 1.0f / (g * g + 0.2f);
#pragma unroll
            for (int s = 0; s < SS; ++s)
                mval[s][kk] = g * kb[(kk * SS + s) * HW + h * 64 + w];
        }
#pragma unroll
        for (int s = 0; s < SS; ++s) {      // softmax over the 9 taps
            float mx = mval[s][0];
#pragma unroll
            for (int kk = 1; kk < KK; ++kk) mx = fmaxf(mx, mval[s][kk]);
            float e[KK], sum = 0.f;
#pragma unroll
            for (int kk = 0; kk < KK; ++kk) { e[kk] = __expf(mval[s][kk] - mx); sum += e[kk]; }
            float inv = 1.0f / sum;
#pragma unroll
            for (int kk = 0; kk < KK; ++kk) smask[kk][s][w] = e[kk] * inv;
        }
    }
    __syncthreads();

    for (int c = ty; c < CIN; c += 4) {
        const float* xb = x + ((size_t)(b * CIN + c)) * HW;
        __builtin_prefetch(xb + 4 * HW, 0, 0);   // next channel group -> global_prefetch_b8
        float v[KK];
#pragma unroll
        for (int kk = 0; kk < KK; ++kk) {
            int dy = kk / 3 - 1, dx = kk % 3 - 1;
            int hh = h + dy, ww = tx + dx;
            v[kk] = ((unsigned)hh < 64u && (unsigned)ww < 64u)
                        ? xb[hh * 64 + ww] : 0.f;
        }
        size_t obase = ((size_t)(b * CIN + c)) * 128 * 128;
#pragma unroll
        for (int p = 0; p < 2; ++p)
#pragma unroll
            for (int q = 0; q < 2; ++q) {
                int s = p * 2 + q;
                float acc = 0.f;
#pragma unroll
                for (int kk = 0; kk < KK; ++kk)
                    acc = fmaf(v[kk], smask[kk][s][tx], acc);
                out[obase + (size_t)(2 * h + p) * 128 + (2 * tx + q)] = acc;
            }
    }
}

// ---------------------------------------------------------------------------
extern "C" void kernel_launch(void* const* d_in, const int* in_sizes, int n_in,
                              void* d_out, int out_size, void* d_ws, size_t ws_size,
                              hipStream_t stream) {
    const float* x  = (const float*)d_in[0];   // [8,256,64,64]
    const float* w1 = (const float*)d_in[1];   // [64,256,1,1]
    const float* b1 = (const float*)d_in[2];   // [64]
    const float* w2 = (const float*)d_in[3];   // [36,64,3,3]
    const float* b2 = (const float*)d_in[4];   // [36]
    float* out = (float*)d_out;                // [8,256,128,128]

    float* ws   = (float*)d_ws;
    float* mean = ws;                          //   32768 floats
    float* comp = ws + 32768;                  // 2097152 floats
    float* kern = comp + (size_t)8 * CCOMP * HW; // 1179648 floats

    gg_mean_kernel <<<128,  256, 0, stream>>>(x, mean);
    gg_comp_gemm   <<<1024, 256, 0, stream>>>(x, w1, b1, comp);
    gg_kern_gemm   <<<2048,  96, 0, stream>>>(comp, w2, b2, kern);
    gg_carafe_kernel<<<512, 256, 0, stream>>>(x, mean, kern, out);
}